// PiecewisePolynomial_45492293599257
// MI455X (gfx1250) — compile-verified
//
#include <hip/hip_runtime.h>
#include <hip/hip_bf16.h>

// Problem constants (from reference)
#define NSEG   128
#define COLS   385            // 3*128 + 1
#define BATCH  32768
#define OUTF   1024
#define MAX_TILES (BATCH/16 + NSEG)   // 2176 (worst-case padded tiles)
#define MAX_SLOTS (MAX_TILES*16)      // 34816

// Workspace layout (int units):
//   [0,128)        cnt[seg]
//   [128,256)      cursor[seg]
//   [256,384)      slot_base[seg]
//   [384,2560)     tile_seg[MAX_TILES]
//   [2560,37376)   perm[MAX_SLOTS]
#define WS_CUR   128
#define WS_SBASE 256
#define WS_TSEG  384
#define WS_PERM  2560
#define WS_INTS  (WS_PERM + MAX_SLOTS)

typedef __attribute__((ext_vector_type(2))) float v2f;
typedef __attribute__((ext_vector_type(8))) float v8f;

__device__ __forceinline__ int seg_of(float x) {
    return (int)floorf((x + 1.0f) * 0.5f * (float)NSEG);
}

// Lagrange basis at Chebyshev-Lobatto nodes X = {-1, -0.5, 0.5, 1}
__device__ __forceinline__ void basis4(float x, int seg, float b[4]) {
    float x_min = (float)seg * (2.0f / NSEG) - 1.0f;
    float xi = (x - x_min) * (float)NSEG - 1.0f;   // 2*(x-xmin)/(2/NSEG) - 1
    float t0 = xi + 1.0f;    // xi - X0
    float t1 = xi + 0.5f;    // xi - X1
    float t2 = xi - 0.5f;    // xi - X2
    float t3 = xi - 1.0f;    // xi - X3
    b[0] = (t1 * -2.0f)          * (t2 * (-2.0f/3.0f)) * (t3 * -0.5f);
    b[1] = (t0 *  2.0f)          * (t2 * -1.0f)        * (t3 * (-2.0f/3.0f));
    b[2] = (t0 * (2.0f/3.0f))    * (t1 *  1.0f)        * (t3 * -2.0f);
    b[3] = (t0 *  0.5f)          * (t1 * (2.0f/3.0f))  * (t2 *  2.0f);
}

// ---- pass 0: zero counters, poison perm with -1 ----
__global__ void k_init(int* __restrict__ ws) {
    int i = blockIdx.x * 256 + threadIdx.x;
    if (i < 256) ws[i] = 0;                 // cnt + cursor
    if (i < MAX_SLOTS) ws[WS_PERM + i] = -1;
}

// ---- pass 1: histogram rows by segment ----
__global__ void k_count(const float* __restrict__ x, int* __restrict__ ws) {
    int b = blockIdx.x * 256 + threadIdx.x;
    if (b < BATCH) atomicAdd(&ws[seg_of(x[b])], 1);
}

// ---- pass 2: scan (pad buckets to multiples of 16), tile->seg map ----
__global__ void k_scan(int* __restrict__ ws) {
    __shared__ int tb[NSEG + 1];
    if (threadIdx.x == 0) {
        int acc = 0;
        for (int s = 0; s < NSEG; ++s) { tb[s] = acc; acc += (ws[s] + 15) >> 4; }
        tb[NSEG] = acc;
    }
    __syncthreads();
    int s = threadIdx.x;            // blockDim.x == 128
    ws[WS_SBASE + s] = tb[s] * 16;
    for (int t = tb[s]; t < tb[s + 1]; ++t) ws[WS_TSEG + t] = s;
    for (int t = tb[NSEG] + threadIdx.x; t < MAX_TILES; t += 128) ws[WS_TSEG + t] = -1;
}

// ---- pass 3: scatter row indices into bucket-contiguous slots ----
// Buckets fill front-to-back, so valid rows form a PREFIX of every tile.
__global__ void k_scatter(const float* __restrict__ x, int* __restrict__ ws) {
    int b = blockIdx.x * 256 + threadIdx.x;
    if (b >= BATCH) return;
    int s = seg_of(x[b]);
    int pos = atomicAdd(&ws[WS_CUR + s], 1);
    ws[WS_PERM + ws[WS_SBASE + s] + pos] = b;
}

// ---- pass 4: one v_wmma_f32_16x16x4_f32 per 16x16 output tile ----
__global__ void __launch_bounds__(256) k_wmma(const float* __restrict__ x,
                                              const float* __restrict__ w,
                                              float* __restrict__ out,
                                              const int* __restrict__ ws) {
    int t   = blockIdx.x;
    int seg = ws[WS_TSEG + t];
    if (seg < 0) return;                 // uniform exit: unused padded tile
    int wid  = seg * 3;
    unsigned lane = threadIdx.x & 31;
    unsigned m    = lane & 15;
    unsigned hi   = lane >> 4;
    unsigned wave = threadIdx.x >> 5;

    // Lane m holds this tile's row M=m (lanes 16-31 mirror 0-15)
    int prow = ws[WS_PERM + t * 16 + (int)m];

    // A matrix 16x4 f32: lane holds row M=m, K pair {2*hi, 2*hi+1}
    v2f a;
    if (prow >= 0) {
        float bb[4];
        basis4(x[prow], seg, bb);
        a.x = bb[2 * hi];
        a.y = bb[2 * hi + 1];
    } else {
        a.x = 0.0f; a.y = 0.0f;          // padded row -> zero contribution
    }

    // Hoist the row permutation out of the o-loop: 8 bpermutes total.
    // D layout: VGPR r -> (M=r, N=lane) lanes 0-15; (M=r+8, N=lane-16) lanes 16-31
    // Store offset for (r,i): rv[r]*OUTF + wave*16 + m  (+ 128*i, a constant
    // per unrolled i -> folds into the store's 24-bit immediate offset).
    int rv[8];
    unsigned uoff[8];
    #pragma unroll
    for (int r = 0; r < 8; ++r) {
        rv[r]   = __shfl(prow, (int)(hi ? (r + 8) : r), 32);
        uoff[r] = (unsigned)rv[r] * OUTF + wave * 16 + m;   // used only if rv>=0
    }
    // Valid rows are a prefix -> tile is full iff all 16 perm entries valid.
    bool full = ((unsigned)(__ballot(prow >= 0) & 0xFFFFull) == 0xFFFFu);

    // Preload all 8 B fragments (4x16 f32: lane = col N=m, K pair {2*hi, 2*hi+1}).
    // Per-i delta is 128*COLS elements (197120 B) -> fits the imm offset, so all
    // 8 loads share one address register.
    const float* wbase = w + (size_t)(wave * 16 + m) * COLS + wid + 2 * hi;
    v2f bm[8];
    #pragma unroll
    for (int i = 0; i < 8; ++i) {
        v2f b0;
        b0.x = wbase[i * 128 * COLS];
        b0.y = wbase[i * 128 * COLS + 1];
        bm[i] = b0;
    }

    // 8 waves x 8 unrolled wmmas cover all 64 output tiles of 16
    #pragma unroll
    for (int i = 0; i < 8; ++i) {
        v8f c = {};
        c = __builtin_amdgcn_wmma_f32_16x16x4_f32(
                false, a, false, bm[i], (short)0, c, false, false);
        if (full) {
            #pragma unroll
            for (int r = 0; r < 8; ++r) out[uoff[r] + 128u * i] = c[r];
        } else {
            #pragma unroll
            for (int r = 0; r < 8; ++r) if (rv[r] >= 0) out[uoff[r] + 128u * i] = c[r];
        }
    }
}

// ---- fallback (ws too small): plain store-bound gather kernel ----
__global__ void k_fallback(const float* __restrict__ x, const float* __restrict__ w,
                           float* __restrict__ out) {
    int idx = blockIdx.x * 256 + threadIdx.x;        // BATCH*OUTF/4 threads
    int b = idx >> 8;                                // 256 float4 groups per row
    int o = (idx & 255) << 2;
    if (b >= BATCH) return;
    float xv = x[b];
    int seg = seg_of(xv);
    int wid = seg * 3;
    float bb[4];
    basis4(xv, seg, bb);
    float4 r;
    float* rp = &r.x;
    #pragma unroll
    for (int j = 0; j < 4; ++j) {
        const float* wp = w + (size_t)(o + j) * COLS + wid;
        rp[j] = bb[0]*wp[0] + bb[1]*wp[1] + bb[2]*wp[2] + bb[3]*wp[3];
    }
    *(float4*)(out + (size_t)b * OUTF + o) = r;
}

extern "C" void kernel_launch(void* const* d_in, const int* in_sizes, int n_in,
                              void* d_out, int out_size, void* d_ws, size_t ws_size,
                              hipStream_t stream) {
    const float* x = (const float*)d_in[0];
    const float* w = (const float*)d_in[1];
    float* out = (float*)d_out;

    size_t need = (size_t)WS_INTS * sizeof(int);
    if (ws_size < need) {
        int total = (BATCH * OUTF) / 4;
        k_fallback<<<(total + 255) / 256, 256, 0, stream>>>(x, w, out);
        return;
    }
    int* ws = (int*)d_ws;
    k_init   <<<MAX_SLOTS / 256, 256, 0, stream>>>(ws);
    k_count  <<<BATCH / 256,     256, 0, stream>>>(x, ws);
    k_scan   <<<1,               128, 0, stream>>>(ws);
    k_scatter<<<BATCH / 256,     256, 0, stream>>>(x, ws);
    k_wmma   <<<MAX_TILES,       256, 0, stream>>>(x, w, out, ws);
}